// HieraFormer_46248207843379
// MI455X (gfx1250) — compile-verified
//
#include <hip/hip_runtime.h>
#include <math.h>

#define D_MODEL 384
#define NHEADS  3
#define DK      128
#define SEQ     2048
#define BATCH   2
#define BH      (BATCH*NHEADS)
#define LN_EPS  1e-6f
// z = energy/2 = (q.k)/(2*sqrt(128))
#define ZSCALE  0.044194173824159216f
#define ES_STRIDE 2052   // 2048 + 4 pad -> conflict-free LDS banks
#define QS_STRIDE 132    // 128 + 4 pad

typedef float v2f __attribute__((ext_vector_type(2)));
typedef float v8f __attribute__((ext_vector_type(8)));

__device__ __forceinline__ v8f wmma_f32_16x16x4(v2f a, v2f b, v8f c) {
    // 8 args: (neg_a, A, neg_b, B, c_mod, C, reuse_a, reuse_b)
    return __builtin_amdgcn_wmma_f32_16x16x4_f32(false, a, false, b, (short)0, c, false, false);
}

__device__ __forceinline__ float waveAllSum(float v) {
#pragma unroll
    for (int m = 16; m >= 1; m >>= 1) v += __shfl_xor(v, m, 32);
    return v;
}
__device__ __forceinline__ float waveAllMax(float v) {
#pragma unroll
    for (int m = 16; m >= 1; m >>= 1) v = fmaxf(v, __shfl_xor(v, m, 32));
    return v;
}

// ---------------------------------------------------------------------------
// Kernel 1: fused QKV projection.  y[b,n,o] = sum_c x[b,c,n] * w[o,c] + bias[o]
// One wave per 16(n) x 16(o) tile; A (from x) shared across the 3 matrices.
// ---------------------------------------------------------------------------
__global__ void qkv_kernel(const float* __restrict__ x,
                           const float* __restrict__ wq, const float* __restrict__ bq,
                           const float* __restrict__ wk, const float* __restrict__ bk,
                           const float* __restrict__ wv, const float* __restrict__ bv,
                           float* __restrict__ qO, float* __restrict__ kO,
                           float* __restrict__ vO) {
    const int lane = threadIdx.x & 31;
    const int wave = threadIdx.x >> 5;
    const int t  = blockIdx.x * 8 + wave;     // 6144 tiles = 2 * 128 * 24
    const int b  = t / (128 * 24);
    const int r  = t % (128 * 24);
    const int n0 = (r / 24) * 16;
    const int o0 = (r % 24) * 16;
    const int hi  = lane >> 4;                // half-wave select (K pairs 0,1 vs 2,3)
    const int l15 = lane & 15;
    const int nn  = n0 + l15;                 // A row  (M)
    const int oo  = o0 + l15;                 // B col  (N)

    v8f aq = {}, ak = {}, av = {};
    const float* xb = x + (size_t)b * D_MODEL * SEQ;
    for (int c = 0; c < D_MODEL; c += 4) {
        const int cA = c + hi * 2;
        v2f a;                                // A[M=nn][K=cA,cA+1] = x[cA][nn]
        a.x = xb[(size_t)(cA    ) * SEQ + nn];
        a.y = xb[(size_t)(cA + 1) * SEQ + nn];
        const v2f bqm = *(const v2f*)(wq + (size_t)oo * D_MODEL + cA);
        const v2f bkm = *(const v2f*)(wk + (size_t)oo * D_MODEL + cA);
        const v2f bvm = *(const v2f*)(wv + (size_t)oo * D_MODEL + cA);
        aq = wmma_f32_16x16x4(a, bqm, aq);
        ak = wmma_f32_16x16x4(a, bkm, ak);
        av = wmma_f32_16x16x4(a, bvm, av);
    }
    const float biq = bq[oo], bik = bk[oo], biv = bv[oo];
    const int h = oo >> 7, d = oo & 127;
#pragma unroll
    for (int rr = 0; rr < 8; ++rr) {
        const int n = n0 + rr + hi * 8;       // D layout: vgpr rr -> M=rr / rr+8
        const size_t idx = ((size_t)(b * NHEADS + h) * SEQ + n) * DK + d;
        qO[idx] = aq[rr] + biq;
        kO[idx] = ak[rr] + bik;
        vO[idx] = av[rr] + biv;
    }
}

// ---------------------------------------------------------------------------
// Kernel 2: per (b,h, 16-query tile): energy->LDS, entmax1.5 via bisection,
// then P @ V.  512 threads = 16 waves.  Dynamic LDS ~137 KB (big WGP LDS).
// ---------------------------------------------------------------------------
__global__ void attn_kernel(const float* __restrict__ qI, const float* __restrict__ kI,
                            const float* __restrict__ vI, float* __restrict__ outW) {
    extern __shared__ float smem[];
    float* qs = smem;                         // [16][QS_STRIDE]
    float* es = smem + 16 * QS_STRIDE;        // [16][ES_STRIDE]

    const int lane = threadIdx.x & 31;
    const int wave = threadIdx.x >> 5;
    const int qt = blockIdx.x;                // 0..127
    const int bh = blockIdx.y;                // 0..5
    const int q0 = qt * 16;
    const float* qp = qI + ((size_t)bh * SEQ + q0) * DK;
    const float* kp = kI + (size_t)bh * SEQ * DK;
    const float* vp = vI + (size_t)bh * SEQ * DK;

    // stage A: q tile -> LDS (padded)
    for (int i = threadIdx.x; i < 16 * DK; i += blockDim.x) {
        const int rr = i >> 7, cc = i & 127;
        qs[rr * QS_STRIDE + cc] = qp[(size_t)rr * DK + cc];
    }
    __syncthreads();

    const int hi  = lane >> 4;
    const int l15 = lane & 15;

    // stage B: z = (q.k)/(2*sqrt(dk)) -> es.  Wave w owns k-tiles w, w+16, ...
    for (int i = 0; i < 8; ++i) {
        const int k0 = (wave + 16 * i) * 16;
        v8f acc = {};
        const float* krow = kp + (size_t)(k0 + l15) * DK + hi * 2;  // B[K][N]=k[k0+N][K]
        const float* qrow = qs + l15 * QS_STRIDE + hi * 2;          // A[M][K]=q[M][K]
        for (int c = 0; c < DK; c += 4) {
            const v2f a  = *(const v2f*)(qrow + c);
            const v2f bm = *(const v2f*)(krow + c);
            acc = wmma_f32_16x16x4(a, bm, acc);
        }
#pragma unroll
        for (int rr = 0; rr < 8; ++rr)
            es[(rr + hi * 8) * ES_STRIDE + k0 + l15] = acc[rr] * ZSCALE;
    }
    __syncthreads();

    // stage C: exact 1.5-entmax per row via bisection on tau.
    // p_j = [z_j - tau]_+^2 with sum p = 1; tau in [max-1, max], f monotone.
    {
        const int row = wave;                 // 16 waves, 16 rows
        float z[64];
#pragma unroll
        for (int j = 0; j < 64; ++j) z[j] = es[row * ES_STRIDE + lane + 32 * j];
        float m = -1e30f;
#pragma unroll
        for (int j = 0; j < 64; ++j) m = fmaxf(m, z[j]);
        m = waveAllMax(m);
        float lo = m - 1.0f, up = m;
        for (int it = 0; it < 40; ++it) {
            const float mid = 0.5f * (lo + up);
            float s = 0.0f;
#pragma unroll
            for (int j = 0; j < 64; ++j) {
                const float tt = fmaxf(z[j] - mid, 0.0f);
                s = fmaf(tt, tt, s);
            }
            s = waveAllSum(s);
            if (s >= 1.0f) lo = mid; else up = mid;
        }
        const float tau = 0.5f * (lo + up);
        float s = 0.0f;
#pragma unroll
        for (int j = 0; j < 64; ++j) {
            const float tt = fmaxf(z[j] - tau, 0.0f);
            z[j] = tt * tt;
            s += z[j];
        }
        s = waveAllSum(s);
        const float inv = 1.0f / s;           // exact-sum normalization
#pragma unroll
        for (int j = 0; j < 64; ++j) es[row * ES_STRIDE + lane + 32 * j] = z[j] * inv;
    }
    __syncthreads();

    // stage D: out = P @ V.  Waves 0..7 each own one 16-wide d-tile, K=2048.
    if (wave < 8) {
        const int d0 = wave * 16;
        v8f acc = {};
        const float* prow = es + l15 * ES_STRIDE + hi * 2;           // A[M][K]=p[M][K]
        const float* vcol = vp + d0 + l15 + (size_t)(hi * 2) * DK;   // B[K][N]=v[K][d0+N]
        for (int k = 0; k < SEQ; k += 4) {
            const v2f a = *(const v2f*)(prow + k);
            v2f bm;
            bm.x = vcol[(size_t)k * DK];
            bm.y = vcol[(size_t)k * DK + DK];
            acc = wmma_f32_16x16x4(a, bm, acc);
        }
        const int b = bh / NHEADS, h = bh % NHEADS;
        const int cidx = h * DK + d0 + l15;
#pragma unroll
        for (int rr = 0; rr < 8; ++rr) {
            const int n = q0 + rr + hi * 8;
            outW[((size_t)b * SEQ + n) * D_MODEL + cidx] = acc[rr];
        }
    }
}

// ---------------------------------------------------------------------------
// Kernel 3: out = x1 + LayerNorm(attn_out).  One wave per (b,n) row of 384.
// Torch semantics: unbiased var (n-1), eps added to std.
// ---------------------------------------------------------------------------
__global__ void lnres_kernel(const float* __restrict__ x, const float* __restrict__ outW,
                             const float* __restrict__ ga, const float* __restrict__ gb,
                             float* __restrict__ out) {
    const int lane = threadIdx.x & 31;
    const int wave = threadIdx.x >> 5;
    const int row  = blockIdx.x * 8 + wave;   // 4096 rows
    const int b = row >> 11, n = row & 2047;
    const float* orow = outW + (size_t)row * D_MODEL;

    float v[12];
    float s = 0.0f;
#pragma unroll
    for (int j = 0; j < 12; ++j) { v[j] = orow[lane + 32 * j]; s += v[j]; }
    s = waveAllSum(s);
    const float mean = s * (1.0f / (float)D_MODEL);
    float ss = 0.0f;
#pragma unroll
    for (int j = 0; j < 12; ++j) { const float t = v[j] - mean; ss = fmaf(t, t, ss); }
    ss = waveAllSum(ss);
    const float stdv = sqrtf(ss * (1.0f / (float)(D_MODEL - 1)));
    const float rs = 1.0f / (stdv + LN_EPS);
    const float* xb = x + (size_t)b * D_MODEL * SEQ + n;
#pragma unroll
    for (int j = 0; j < 12; ++j) {
        const int c = lane + 32 * j;
        out[(size_t)row * D_MODEL + c] =
            xb[(size_t)c * SEQ] + fmaf(ga[c] * (v[j] - mean), rs, gb[c]);
    }
}

// ---------------------------------------------------------------------------
extern "C" void kernel_launch(void* const* d_in, const int* in_sizes, int n_in,
                              void* d_out, int out_size, void* d_ws, size_t ws_size,
                              hipStream_t stream) {
    const float* x  = (const float*)d_in[0];
    const float* wq = (const float*)d_in[1];
    const float* bq = (const float*)d_in[2];
    const float* wk = (const float*)d_in[3];
    const float* bk = (const float*)d_in[4];
    const float* wv = (const float*)d_in[5];
    const float* bv = (const float*)d_in[6];
    const float* ga = (const float*)d_in[7];
    const float* gb = (const float*)d_in[8];
    float* out = (float*)d_out;

    float* ws = (float*)d_ws;
    const size_t qkvElems = (size_t)BH * SEQ * DK;     // 1,572,864 floats each
    float* qW = ws;
    float* kW = qW + qkvElems;
    float* vW = kW + qkvElems;
    float* oW = vW + qkvElems;                         // B*N*C floats

    qkv_kernel<<<768, 256, 0, stream>>>(x, wq, bq, wk, bk, wv, bv, qW, kW, vW);

    dim3 g2(SEQ / 16, BH);                             // 128 x 6
    const size_t shmem = (size_t)(16 * QS_STRIDE + 16 * ES_STRIDE) * sizeof(float);
    attn_kernel<<<g2, 512, shmem, stream>>>(qW, kW, vW, oW);

    lnres_kernel<<<512, 256, 0, stream>>>(x, oW, ga, gb, out);
}